// DiffJPEG_79628693668437
// MI455X (gfx1250) — compile-verified
//
#include <hip/hip_runtime.h>
#include <math.h>

// ---------------------------------------------------------------------------
// DiffJPEG fused kernel for gfx1250 (MI455X), fp32 WMMA path, v3.
//
// One wave32 owns a horizontal strip of 4 16x16 pixel tiles. Per tile:
//   - RGB loaded directly in WMMA B-fragment lane layout, converted to YCbCr.
//   - DCT/IDCT of the four 8x8 Y blocks as 16x16x16 matmuls with the
//     block-diagonal cosine matrix G = diag(C,C): 4 chained
//     V_WMMA_F32_16X16X4_F32 per matmul, 16 per chain, 32 per tile.
//   - Cb/Cr pooled 8x8 blocks packed block-diagonally -> one chroma chain.
// LDS staging orientation chosen so every fragment read is one contiguous
// 8-byte-aligned pair (ds_load_b64). All divisions/cosf hoisted to per-wave
// setup (amortized over 4 tiles). v3: branch-free chroma pooling (the Cb/Cr
// region split is compile-time per k, lane validity is one cndmask), and
// upsample processes row-pairs so each pooled chroma texel is read once.
// ---------------------------------------------------------------------------

typedef __attribute__((ext_vector_type(2))) float v2f;
typedef __attribute__((ext_vector_type(8))) float v8f;

// Compiler-level fence: LDS ops within a wave execute in order (DScnt, ISA
// 7.3), so intra-wave cross-lane LDS traffic only needs compile-time ordering.
#define LDS_FENCE() __asm__ __volatile__("" ::: "memory")

__device__ __forceinline__ v8f wmma4(v2f a, v2f b, v8f c) {
  // D = A(16x4) * B(4x16) + C ; f32 in/out.
  return __builtin_amdgcn_wmma_f32_16x16x4_f32(false, a, false, b, (short)0, c,
                                               false, false);
}

// Raw JPEG luma table as written in the reference (before the .T):
// reference table[u][v] = YRAW[v][u].
__constant__ float YRAW[8][8] = {
    {16, 11, 10, 16, 24, 40, 51, 61},    {12, 12, 14, 19, 26, 58, 60, 55},
    {14, 13, 16, 24, 40, 57, 69, 56},    {14, 17, 22, 29, 51, 87, 80, 62},
    {18, 22, 37, 56, 68, 109, 103, 77},  {24, 35, 55, 64, 81, 104, 113, 92},
    {49, 64, 78, 87, 103, 121, 120, 101},{72, 92, 95, 98, 112, 100, 103, 99}};

// Chroma 4x4 corner (symmetric => .T is a no-op); rest of the 8x8 is 99.
__constant__ float CRAW4[4][4] = {
    {17, 18, 24, 47}, {18, 21, 26, 66}, {24, 26, 56, 99}, {47, 66, 99, 99}};

#define QFACTOR 0.4f  // QUALITY=80 -> (200-160)/100
#define RSQRT2  0.70710678118654752f
#define PI_16   0.19634954084936207f
#define LSTR    18    // LDS row stride (even => b64-aligned pairs; low conflict)

// G = diag(C, C), C[x][u] = cos((2x+1) u pi/16)
__device__ __forceinline__ float Gval(int i, int j) {
  if ((i >> 3) != (j >> 3)) return 0.0f;
  return cosf((float)((2 * (i & 7) + 1) * (j & 7)) * PI_16);
}

// bIn: B-fragments of (X - 128). vout: rec = 0.25*G*quant(Gt*X*G)*Gt + 128 in
// WMMA C/D register layout ((lane l, slot r) <-> (M = r+8*(l>>4), N = l&15)).
// m1[r] = 0.25*s/tab, m2[r] = tab*s with s = alpha_u*alpha_v at (r, lo&7).
__device__ __forceinline__ void chain(const v2f* bIn, float* vout, float* S,
                                      const v2f* F1, const v2f* F2,
                                      const float* m1, const float* m2,
                                      int lo, int hi) {
  const int ro = 8 * hi;
  const int ldA = lo * LSTR + 2 * hi;  // shared base for all pair loads
  const v8f vzero = {0.f, 0.f, 0.f, 0.f, 0.f, 0.f, 0.f, 0.f};

  // M1: T1 = Gt * X (A = F1 const frags, B = input frags; no LDS round-trip)
  v8f acc = vzero;
#pragma unroll
  for (int k = 0; k < 4; ++k) acc = wmma4(F1[k], bIn[k], acc);

  // stage T1 row-major (consumed as A-frags: contiguous pairs along a row)
  LDS_FENCE();
#pragma unroll
  for (int r = 0; r < 8; ++r) S[(r + ro) * LSTR + lo] = acc[r];
  LDS_FENCE();

  // M2: coef0 = T1 * G (A = T1 from LDS, B = F1)
  v8f acc2 = vzero;
#pragma unroll
  for (int k = 0; k < 4; ++k) {
    v2f a = {S[ldA + 4 * k], S[ldA + 4 * k + 1]};
    acc2 = wmma4(a, F1[k], acc2);
  }

  // differentiable quantization; stage e TRANSPOSED so M3's B-frags are
  // contiguous pairs at the same addresses as the A-frag reads.
  LDS_FENCE();
#pragma unroll
  for (int r = 0; r < 8; ++r) {
    const float t = acc2[r] * m1[r];
    const float rn = rintf(t);  // round-half-even, matches jnp.round
    const float d = t - rn;
    S[lo * LSTR + ro + r] = (rn + d * d * d) * m2[r];
  }
  LDS_FENCE();

  // M3: T2 = G * e (A = F2, B = e from LDS)
  acc = vzero;
#pragma unroll
  for (int k = 0; k < 4; ++k) {
    v2f b = {S[ldA + 4 * k], S[ldA + 4 * k + 1]};
    acc = wmma4(F2[k], b, acc);
  }

  // stage T2 row-major
  LDS_FENCE();
#pragma unroll
  for (int r = 0; r < 8; ++r) S[(r + ro) * LSTR + lo] = acc[r];
  LDS_FENCE();

  // M4: rec0 = T2 * Gt (A = T2 from LDS, B = F2)
  acc2 = vzero;
#pragma unroll
  for (int k = 0; k < 4; ++k) {
    v2f a = {S[ldA + 4 * k], S[ldA + 4 * k + 1]};
    acc2 = wmma4(a, F2[k], acc2);
  }
#pragma unroll
  for (int r = 0; r < 8; ++r) vout[r] = 0.25f * acc2[r] + 128.0f;
}

__global__ __launch_bounds__(256) void diffjpeg_kernel(
    const float* __restrict__ x, float* __restrict__ out) {
  __shared__ float smem[8 * 3 * (16 * LSTR)];

  const int tidb = threadIdx.x;
  const int wv = tidb >> 5;
  const int lane = tidb & 31;
  const int lo = lane & 15;
  const int hi = lane >> 4;
  const int h2 = 2 * hi;

  float* S = smem + wv * (3 * 16 * LSTR);
  float* bufA = S + 16 * LSTR;   // full-res Cb, later reconstructed chroma
  float* bufB = S + 32 * LSTR;   // full-res Cr

  // ---- per-wave setup (amortized over 4 tiles) -----------------------------
  // Constant G fragments: F1 = A-frag(Gt) == B-frag(G); F2 = A-frag(G) ==
  // B-frag(Gt).
  v2f F1[4], F2[4];
#pragma unroll
  for (int k = 0; k < 4; ++k) {
    const int j0 = 4 * k + h2;
    F1[k].x = Gval(j0, lo);
    F1[k].y = Gval(j0 + 1, lo);
    F2[k].x = Gval(lo, j0);
    F2[k].y = Gval(lo, j0 + 1);
  }

  // Quant multipliers at (u%8, v%8) = (r, lo&7): t = raw*m1 ; e = q*m2.
  const int c7 = lane & 7;
  const float av = (c7 == 0) ? RSQRT2 : 1.0f;
  float m1Y[8], m2Y[8], m1C[8], m2C[8];
#pragma unroll
  for (int r = 0; r < 8; ++r) {
    const float s = ((r == 0) ? RSQRT2 : 1.0f) * av;  // alpha2 ; SCALE=0.25*s
    const float ty = YRAW[c7][r] * QFACTOR;           // _Y_TABLE = RAW.T
    const float tc = ((r < 4 && c7 < 4) ? CRAW4[r][c7] : 99.0f) * QFACTOR;
    m1Y[r] = 0.25f * s / ty;  // divisions only here, once per wave
    m2Y[r] = ty * s;
    m1C[r] = 0.25f * s / tc;
    m2C[r] = tc * s;
  }

  // Lane validity for the two diagonal chroma blocks (see pooling below).
  const float vCb = (lo < 8) ? 1.0f : 0.0f;   // Cb block: cols 0..7
  const float vCr = (lo >= 8) ? 1.0f : 0.0f;  // Cr block: cols 8..15

  // Strip of 4 horizontally-adjacent tiles (never crosses row/image bounds).
  const int tile0 = (blockIdx.x * 8 + wv) * 4;
  const int img = tile0 >> 10;
  const int tixb = tile0 & 1023;
  const int row0 = (tixb >> 5) << 4;
  const int colB = (tixb & 31) << 4;

  const unsigned base0 = (unsigned)(img * 3 + 0) << 18;  // 512*512 = 2^18
  const unsigned base1 = (unsigned)(img * 3 + 1) << 18;
  const unsigned base2 = (unsigned)(img * 3 + 2) << 18;

#pragma unroll 1
  for (int t = 0; t < 4; ++t) {
    const int col0 = colB + 16 * t;

    // ---- load RGB in B-fragment layout; convert; Cb/Cr full-res to LDS ----
    LDS_FENCE();
    v2f Yb[4];
#pragma unroll
    for (int k = 0; k < 4; ++k) {
#pragma unroll
      for (int c = 0; c < 2; ++c) {
        const int m = 4 * k + h2 + c;  // tile row owned by this lane/slot
        const unsigned off =
            (unsigned)(row0 + m) * 512u + (unsigned)(col0 + lo);
        const float rr = x[base0 + off] * 255.0f;
        const float gg = x[base1 + off] * 255.0f;
        const float bb = x[base2 + off] * 255.0f;
        const float yv = 0.299f * rr + 0.587f * gg + 0.114f * bb - 128.0f;
        bufA[m * LSTR + lo] =
            -0.168736f * rr - 0.331264f * gg + 0.5f * bb + 128.0f;
        bufB[m * LSTR + lo] =
            0.5f * rr - 0.418688f * gg - 0.081312f * bb + 128.0f;
        if (c == 0) Yb[k].x = yv; else Yb[k].y = yv;
      }
    }
    LDS_FENCE();

    // ---- Y: four 8x8 blocks via block-diagonal G --------------------------
    float recY[8];
    chain(Yb, recY, S, F1, F2, m1Y, m2Y, lo, hi);

    // ---- chroma: branch-free 2x2 pool, build diag(Cb-128, Cr-128) B-frags -
    // m = 4k+2hi+c  =>  m<8 iff k<2 (compile-time): k=0,1 -> Cb, k=2,3 -> Cr.
    // Only lane validity (lo<8 vs lo>=8) is runtime: one multiply by 0/1.
    LDS_FENCE();
    v2f Cf[4];
#pragma unroll
    for (int k = 0; k < 4; ++k) {
      const float* src = (k < 2) ? bufA : bufB;    // compile-time select
      const float vmask = (k < 2) ? vCb : vCr;
      const int c2 = 2 * (lo & 7);
#pragma unroll
      for (int c = 0; c < 2; ++c) {
        const int m = 4 * k + h2 + c;
        const int r2 = 2 * (m & 7);
        const float sum = src[r2 * LSTR + c2] + src[r2 * LSTR + c2 + 1] +
                          src[(r2 + 1) * LSTR + c2] +
                          src[(r2 + 1) * LSTR + c2 + 1];
        const float val = (0.25f * sum - 128.0f) * vmask;  // off-diag -> 0
        if (c == 0) Cf[k].x = val; else Cf[k].y = val;
      }
    }

    float recC[8];
    chain(Cf, recC, S, F1, F2, m1C, m2C, lo, hi);

    // ---- stash reconstructed chroma (Cb TL 8x8, Cr BR 8x8) for upsample ---
    LDS_FENCE();
#pragma unroll
    for (int r = 0; r < 8; ++r) bufA[(r + 8 * hi) * LSTR + lo] = recC[r];
    LDS_FENCE();

    // ---- nearest 2x upsample + YCbCr->RGB + clip + store ------------------
    // Output rows 2p and 2p+1 share one pooled chroma texel: read it once.
    const int nh = lo >> 1;
#pragma unroll
    for (int p = 0; p < 4; ++p) {
      const int mh = p + 4 * hi;  // (m>>1) for both rows of the pair
      const float cb = bufA[mh * LSTR + nh] - 128.0f;
      const float cr = bufA[(8 + mh) * LSTR + (8 + nh)] - 128.0f;
#pragma unroll
      for (int c = 0; c < 2; ++c) {
        const int r = 2 * p + c;
        const int m = r + 8 * hi;
        const float y = recY[r];
        float R = y + 1.402f * cr;
        float G = y - 0.344136f * cb - 0.714136f * cr;
        float B = y + 1.772f * cb;
        R = fminf(fmaxf(R, 0.0f), 255.0f) * (1.0f / 255.0f);
        G = fminf(fmaxf(G, 0.0f), 255.0f) * (1.0f / 255.0f);
        B = fminf(fmaxf(B, 0.0f), 255.0f) * (1.0f / 255.0f);
        const unsigned off =
            (unsigned)(row0 + m) * 512u + (unsigned)(col0 + lo);
        out[base0 + off] = R;
        out[base1 + off] = G;
        out[base2 + off] = B;
      }
    }
  }
}

extern "C" void kernel_launch(void* const* d_in, const int* in_sizes, int n_in,
                              void* d_out, int out_size, void* d_ws,
                              size_t ws_size, hipStream_t stream) {
  (void)in_sizes; (void)n_in; (void)d_ws; (void)ws_size; (void)out_size;
  const float* x = (const float*)d_in[0];
  float* o = (float*)d_out;
  // 65536 tiles / (8 waves * 4 tiles per wave) = 2048 blocks of 256 threads.
  diffjpeg_kernel<<<dim3(2048), dim3(256), 0, stream>>>(x, o);
}